// GaussianPointCloudRasterisation_74904229642909
// MI455X (gfx1250) — compile-verified
//
#include <hip/hip_runtime.h>
#include <math.h>

typedef __attribute__((ext_vector_type(2))) float v2f;
typedef __attribute__((ext_vector_type(8))) float v8f;

// Problem constants (match reference)
static constexpr int   HW      = 256;   // H == W == 256
static constexpr int   NPTS    = 256;
static constexpr float NEARZ   = 0.8f;
static constexpr float FARZ    = 1000.0f;
static constexpr float AMIN    = 1.0f / 255.0f;
static constexpr float ACLAMP  = 0.99f;
static constexpr float TSTOP   = 1e-4f;
static constexpr float MARGIN  = 48.0f;

// Workspace layout (floats):
//   params[256][8] = {u, v, ia, 2*ib, ic, o, 0, 0}   (32B record -> b128+b64 LDS loads)
//   feat16[256][16] = {r, g, b, z, 1, 0...0}          (WMMA B rows)
static constexpr int WS_P = 0, WS_F = NPTS * 8, WS_TOTAL = NPTS * 8 + NPTS * 16;

// ---------------------------------------------------------------------------
// Kernel 1: per-gaussian preprocessing + bitonic depth sort (1 block, 256 thr)
// ---------------------------------------------------------------------------
__global__ __launch_bounds__(256) void gs_preprocess(
    const float* __restrict__ pc, const float* __restrict__ ft,
    const float* __restrict__ Km, const float* __restrict__ Tm,
    float* __restrict__ ws)
{
    __shared__ float sk[NPTS];
    __shared__ int   si[NPTS];
    __shared__ float pu[NPTS], pv[NPTS], pia[NPTS], pib[NPTS], pic[NPTS], po[NPTS];
    __shared__ float pr[NPTS], pg[NPTS], pb[NPTS], pz[NPTS];

    const int i = threadIdx.x;

    float Rw[3][3] = {{Tm[0],Tm[1],Tm[2]},{Tm[4],Tm[5],Tm[6]},{Tm[8],Tm[9],Tm[10]}};
    float tw[3]    = {Tm[3], Tm[7], Tm[11]};
    const float fx = Km[0], cx = Km[2], fy = Km[4], cy = Km[5];

    float p[3] = {pc[3*i+0], pc[3*i+1], pc[3*i+2]};
    float xc[3];
#pragma unroll
    for (int r = 0; r < 3; ++r)
        xc[r] = Rw[r][0]*p[0] + Rw[r][1]*p[1] + Rw[r][2]*p[2] + tw[r];
    float z = xc[2];
    bool valid = (z > NEARZ) && (z < FARZ);
    float zsafe = valid ? z : 1.0f;
    float u = fx * xc[0] / zsafe + cx;
    float v = fy * xc[1] / zsafe + cy;
    valid = valid && (u >= -MARGIN) && (u < (float)HW + MARGIN)
                  && (v >= -MARGIN) && (v < (float)HW + MARGIN);

    const float* f = ft + 56 * i;
    float qx = f[0], qy = f[1], qz = f[2], qw = f[3];
    float inq = 1.0f / sqrtf(qx*qx + qy*qy + qz*qz + qw*qw);
    qx *= inq; qy *= inq; qz *= inq; qw *= inq;
    float sx = __expf(f[4]), sy = __expf(f[5]), sz = __expf(f[6]);
    float s2[3] = { sx*sx, sy*sy, sz*sz };

    float R[3][3] = {
        {1.0f-2.0f*(qy*qy+qz*qz), 2.0f*(qx*qy-qw*qz),      2.0f*(qx*qz+qw*qy)},
        {2.0f*(qx*qy+qw*qz),      1.0f-2.0f*(qx*qx+qz*qz), 2.0f*(qy*qz-qw*qx)},
        {2.0f*(qx*qz-qw*qy),      2.0f*(qy*qz+qw*qx),      1.0f-2.0f*(qx*qx+qy*qy)}};

    // Sigma = R * diag(s^2) * R^T (world), then Sigma_cam = Rw * Sigma * Rw^T
    float Sg[3][3];
#pragma unroll
    for (int r = 0; r < 3; ++r)
#pragma unroll
        for (int c = 0; c < 3; ++c)
            Sg[r][c] = R[r][0]*R[c][0]*s2[0] + R[r][1]*R[c][1]*s2[1] + R[r][2]*R[c][2]*s2[2];

    float Tp[3][3];
#pragma unroll
    for (int r = 0; r < 3; ++r)
#pragma unroll
        for (int c = 0; c < 3; ++c)
            Tp[r][c] = Rw[r][0]*Sg[0][c] + Rw[r][1]*Sg[1][c] + Rw[r][2]*Sg[2][c];
    float Sc[3][3];
#pragma unroll
    for (int r = 0; r < 3; ++r)
#pragma unroll
        for (int c = 0; c < 3; ++c)
            Sc[r][c] = Tp[r][0]*Rw[c][0] + Tp[r][1]*Rw[c][1] + Tp[r][2]*Rw[c][2];

    // J (2x3), cov2d = J * Sigma_cam * J^T
    float iz = 1.0f / zsafe;
    float J0[3] = { fx*iz, 0.0f, -fx*xc[0]*iz*iz };
    float J1[3] = { 0.0f,  fy*iz, -fy*xc[1]*iz*iz };
    float JS0[3], JS1[3];
#pragma unroll
    for (int c = 0; c < 3; ++c) {
        JS0[c] = J0[0]*Sc[0][c] + J0[1]*Sc[1][c] + J0[2]*Sc[2][c];
        JS1[c] = J1[0]*Sc[0][c] + J1[1]*Sc[1][c] + J1[2]*Sc[2][c];
    }
    float ca = JS0[0]*J0[0] + JS0[1]*J0[1] + JS0[2]*J0[2];
    float cb = JS0[0]*J1[0] + JS0[1]*J1[1] + JS0[2]*J1[2];
    float cc = JS1[0]*J1[0] + JS1[1]*J1[1] + JS1[2]*J1[2];

    float det = ca*cc - cb*cb;
    bool dok = det > 1e-12f;
    valid = valid && dok;
    float dsf = dok ? det : 1.0f;
    float ia = cc / dsf, ib = -cb / dsf, ic = ca / dsf;

    // View direction + SH16 color
    float cam[3];
#pragma unroll
    for (int c = 0; c < 3; ++c)
        cam[c] = -(Rw[0][c]*tw[0] + Rw[1][c]*tw[1] + Rw[2][c]*tw[2]);
    float d0 = p[0]-cam[0], d1 = p[1]-cam[1], d2 = p[2]-cam[2];
    float nrm = sqrtf(d0*d0 + d1*d1 + d2*d2) + 1e-8f;
    float dx = d0/nrm, dy = d1/nrm, dz = d2/nrm;
    float xx = dx*dx, yy = dy*dy, zz = dz*dz, xy = dx*dy, yz = dy*dz, xz = dx*dz;
    float Y[16] = {
        0.28209479177387814f,
        -0.4886025119029199f * dy,
         0.4886025119029199f * dz,
        -0.4886025119029199f * dx,
         1.0925484305920792f * xy,
        -1.0925484305920792f * yz,
         0.31539156525252005f * (2.0f*zz - xx - yy),
        -1.0925484305920792f * xz,
         0.5462742152960396f * (xx - yy),
        -0.5900435899266435f * dy * (3.0f*xx - yy),
         2.890611442640554f  * xy * dz,
        -0.4570457994644658f * dy * (4.0f*zz - xx - yy),
         0.3731763325901154f * dz * (2.0f*zz - 3.0f*xx - 3.0f*yy),
        -0.4570457994644658f * dx * (4.0f*zz - xx - yy),
         1.445305721320277f  * dz * (xx - yy),
        -0.5900435899266435f * dx * (xx - 3.0f*yy)};
    float col[3];
#pragma unroll
    for (int ch = 0; ch < 3; ++ch) {
        float s = 0.0f;
#pragma unroll
        for (int k = 0; k < 16; ++k) s += f[8 + ch*16 + k] * Y[k];
        col[ch] = 1.0f / (1.0f + __expf(-s));
    }
    float opa = 1.0f / (1.0f + __expf(-f[7]));

    pu[i] = u;  pv[i] = v;  pia[i] = ia; pib[i] = ib; pic[i] = ic;
    po[i] = valid ? opa : 0.0f;
    pr[i] = col[0]; pg[i] = col[1]; pb[i] = col[2];
    pz[i] = valid ? z : 0.0f;
    sk[i] = valid ? z : __builtin_inff();
    si[i] = i;
    __syncthreads();

    // Bitonic sort (ascending by depth key), 256 elements / 256 threads
    for (int size = 2; size <= NPTS; size <<= 1) {
        for (int stride = size >> 1; stride > 0; stride >>= 1) {
            int pos = i, partner = pos ^ stride;
            if (partner > pos) {
                bool asc = ((pos & size) == 0);
                float kp = sk[pos], kq = sk[partner];
                if ((kp > kq) == asc) {
                    sk[pos] = kq; sk[partner] = kp;
                    int t = si[pos]; si[pos] = si[partner]; si[partner] = t;
                }
            }
            __syncthreads();
        }
    }

    // Scatter sorted records:
    //   params[i] = {u, v, ia, 2*ib, ic, o, 0, 0} ; feat16[i] = {r,g,b,z,1,0..}
    int src = si[i];
    float* pp = ws + WS_P + 8*i;
    pp[0] = pu[src]; pp[1] = pv[src]; pp[2] = pia[src]; pp[3] = 2.0f*pib[src];
    pp[4] = pic[src]; pp[5] = po[src]; pp[6] = 0.0f;   pp[7] = 0.0f;
    float* fo = ws + WS_F + 16*i;
    fo[0] = pr[src]; fo[1] = pg[src]; fo[2] = pb[src]; fo[3] = pz[src]; fo[4] = 1.0f;
#pragma unroll
    for (int k = 5; k < 16; ++k) fo[k] = 0.0f;
}

// ---------------------------------------------------------------------------
// Kernel 2: rasterize. 256 thr = 8 waves; each wave owns 16 pixels.
// Serial transmittance scan in VALU; blend via V_WMMA_F32_16X16X4_F32.
// ---------------------------------------------------------------------------
__global__ __launch_bounds__(256) void gs_rasterize(
    const float* __restrict__ ws, float* __restrict__ out)
{
    __shared__ __align__(16) float s[WS_TOTAL];
    const int tid = threadIdx.x;

    // Vectorized global -> LDS staging (b128 loads + b128 ds stores)
    {
        float4*       s4 = reinterpret_cast<float4*>(s);
        const float4* w4 = reinterpret_cast<const float4*>(ws);
        for (int idx = tid; idx < WS_TOTAL / 4; idx += 256) s4[idx] = w4[idx];
    }
    __syncthreads();

    const int lane = tid & 31;
    const int wave = tid >> 5;
    const bool hi  = (lane & 16) != 0;      // loop-invariant A/B half selector
    const int colc = lane & 15;
    const int pixBase = blockIdx.x * 128 + wave * 16;
    const int myPix   = pixBase + colc;     // lanes L and L+16 track the same pixel
    const float px = (float)(myPix & (HW - 1)) + 0.5f;
    const float py = (float)(myPix >> 8) + 0.5f;

    // Hoisted B-matrix base: this lane reads rows (k0 + 2*half + {0,1}), col colc
    const float* bptr = s + WS_F + (hi ? 32 : 0) + colc;
    const int accwLane = (lane & 16) | 4;   // channel-4 lane of this half

    v8f acc = {0.f,0.f,0.f,0.f,0.f,0.f,0.f,0.f};
    float T = 1.0f;
    bool broken = false;

    for (int k0 = 0; k0 < NPTS; k0 += 4) {
        float w0, w1, w2, w3;
#pragma unroll
        for (int j = 0; j < 4; ++j) {
            int k = k0 + j;
            // one ds_load_b128 + one ds_load_b64 per gaussian
            float4 g0 = *reinterpret_cast<const float4*>(s + WS_P + 8*k);      // u v ia 2ib
            float2 g1 = *reinterpret_cast<const float2*>(s + WS_P + 8*k + 4);  // ic o
            float du = px - g0.x;
            float dv = py - g0.y;
            float e = -0.5f * (g0.z*du*du + g0.w*du*dv + g1.x*dv*dv);
            e = fminf(e, 0.0f);
            float al = __expf(e) * g1.y;
            bool eff = (al >= AMIN);
            float a  = eff ? fminf(al, ACLAMP) : 0.0f;
            float Tn = T * (1.0f - a);
            bool brk = eff && (Tn <= TSTOP);
            float wj = (broken || brk) ? 0.0f : a * T;
            broken = broken || brk;
            T = Tn;
            if (j == 0) w0 = wj; else if (j == 1) w1 = wj;
            else if (j == 2) w2 = wj; else w3 = wj;
        }
        // A(16x4): lanes 0-15 hold K=0,1 ; lanes 16-31 hold K=2,3 (same rows)
        v2f A; A.x = hi ? w2 : w0; A.y = hi ? w3 : w1;
        // B(4x16): lanes 0-15 rows K=0,1 ; lanes 16-31 rows K=2,3
        v2f B; B.x = bptr[k0 * 16]; B.y = bptr[k0 * 16 + 16];
        acc = __builtin_amdgcn_wmma_f32_16x16x4_f32(
                  false, A, false, B, (short)0, acc, false, false);
        if (__all((int)broken)) break;   // wave-uniform early termination
    }

    // D(16x16): vgpr r, lane -> pixel = pixBase + r + 8*half, channel = colc
#pragma unroll
    for (int r = 0; r < 8; ++r) {
        float val = acc[r];
        float accw = __shfl(val, accwLane, 32);  // channel-4 (acc) value
        int pix = pixBase + r + (hi ? 8 : 0);
        if (colc < 3)        out[pix*3 + colc] = val;                       // image
        else if (colc == 3)  out[3*HW*HW + pix] = val / fmaxf(accw, 1e-6f); // depth
        else if (colc == 4)  out[4*HW*HW + pix] = val;                      // acc
    }
}

extern "C" void kernel_launch(void* const* d_in, const int* in_sizes, int n_in,
                              void* d_out, int out_size, void* d_ws, size_t ws_size,
                              hipStream_t stream) {
    const float* pc = (const float*)d_in[0];   // (N,3)
    const float* ft = (const float*)d_in[1];   // (N,56)
    const float* Km = (const float*)d_in[2];   // (3,3)
    const float* Tm = (const float*)d_in[3];   // (4,4)
    float* out = (float*)d_out;                // image(256*256*3) ++ depth ++ acc
    float* ws  = (float*)d_ws;                 // WS_TOTAL floats

    gs_preprocess<<<1, 256, 0, stream>>>(pc, ft, Km, Tm, ws);
    gs_rasterize<<<(HW*HW)/128, 256, 0, stream>>>(ws, out);
}